// GraphEmbedder_87771951661882
// MI455X (gfx1250) — compile-verified
//
#include <hip/hip_runtime.h>

#define BB 8
#define NN 1024
#define EE 10
#define EP 16   // E padded to 16 for WMMA tiles

typedef __attribute__((ext_vector_type(2))) float v2f;
typedef __attribute__((ext_vector_type(8))) float v8f;

// ---------------------------------------------------------------------------
// Kernel 1: S[b,j,e] = sum_i relu(weights[b,i,j]*nbwew_w[e]+nbwew_b[e])
//           base[b,j,o] = feat*sel + sel_b + nbw(S) + nbw_b     (padded to 16)
//           emb1[b,j,o] = relu(base + nbp_b)   (ITERS iteration #1, emb0 == 0)
// Block = 32 columns x 8 row-chunks (one wave per chunk): every wave issues a
// fully coalesced 128 B read of `weights` per step; the 8 partial sums are
// reduced through LDS.  grid = (N/32, B) = 256 blocks -> enough concurrency
// to stream 32 MB at HBM rate.
// ---------------------------------------------------------------------------
__global__ void k_edge_base(const float* __restrict__ features,
                            const float* __restrict__ weights,
                            const float* __restrict__ sel_w,  const float* __restrict__ sel_b,
                            const float* __restrict__ nbw_w,  const float* __restrict__ nbw_b,
                            const float* __restrict__ nbwew_w,const float* __restrict__ nbwew_b,
                            const float* __restrict__ nbp_b,
                            float* __restrict__ base, float* __restrict__ emb1)
{
    __shared__ float sAcc[8 * 32 * EE];                    // 10 KB

    const int col   = threadIdx.x & 31;                    // column within block
    const int chunk = threadIdx.x >> 5;                    // which i-chunk (== wave)
    const int j = blockIdx.x * 32 + col;                   // dest node (column)
    const int b = blockIdx.y;

    float nw[EE], nb[EE], acc[EE];
    #pragma unroll
    for (int e = 0; e < EE; ++e) { nw[e] = nbwew_w[e]; nb[e] = nbwew_b[e]; acc[e] = 0.f; }

    const float* wcol = weights + (size_t)b * NN * NN + j;
    const int ibeg = chunk * (NN / 8);
    for (int i = ibeg; i < ibeg + NN / 8; ++i) {
        float w = wcol[(size_t)i * NN];
        #pragma unroll
        for (int e = 0; e < EE; ++e) {
            float t = fmaf(w, nw[e], nb[e]);
            acc[e] += (t > 0.f) ? t : 0.f;
        }
    }
    #pragma unroll
    for (int e = 0; e < EE; ++e)
        sAcc[(chunk * 32 + col) * EE + e] = acc[e];
    __syncthreads();

    if (threadIdx.x < 32) {                                // one thread per column
        float s[EE];
        #pragma unroll
        for (int e = 0; e < EE; ++e) s[e] = 0.f;
        for (int c = 0; c < 8; ++c)
            #pragma unroll
            for (int e = 0; e < EE; ++e) s[e] += sAcc[(c * 32 + col) * EE + e];

        const float f = features[b * NN + j];
        const size_t o0 = ((size_t)b * NN + j) * EP;
        #pragma unroll
        for (int o = 0; o < EP; ++o) {
            float v = 0.f, em = 0.f;
            if (o < EE) {
                v = fmaf(f, sel_w[o], sel_b[o]) + nbw_b[o];
                #pragma unroll
                for (int e = 0; e < EE; ++e) v = fmaf(s[e], nbw_w[o * EE + e], v);
                em = v + nbp_b[o];
                em = (em > 0.f) ? em : 0.f;
            }
            base[o0 + o] = v;                              // padded cols 10..15 stay 0
            emb1[o0 + o] = em;
        }
    }
}

// ---------------------------------------------------------------------------
// Kernel 2: one propagation step  emb_out = relu(base + nbp(A . emb_in))
// Each block owns ONE 16x16 output tile; its 8 waves split K=1024 into
// 128-wide chunks and each runs 32 V_WMMA_F32_16X16X4_F32 accumulations
// (A = adjacency 16x4 tile, B = emb_in 4x16 tile, E padded to 16).  The 8
// partial f32 tiles are summed through LDS, then the 10x10 nbp GEMM +
// bias + relu epilogue runs one element per thread.
// grid = (N/16, B) = 512 blocks -> 8x the concurrency of a tile-per-wave
// scheme with identical HBM traffic.
// ---------------------------------------------------------------------------
__global__ void k_propagate(const float* __restrict__ adj,
                            const float* __restrict__ base,
                            const float* __restrict__ nbp_w, const float* __restrict__ nbp_b,
                            const float* __restrict__ emb_in,
                            float* __restrict__ emb_out)
{
    __shared__ float sP[8 * 256];                  // 8 partial 16x16 tiles, 8 KB

    const int b     = blockIdx.y;
    const int lane  = threadIdx.x & 31;
    const int wave  = threadIdx.x >> 5;
    const int col   = lane & 15;                   // N-index of B/C/D, M-index of A
    const int khalf = lane >> 4;                   // 0 or 1 -> which K pair
    const int i0    = blockIdx.x * 16;

    const float* arow = adj    + ((size_t)b * NN + (i0 + col)) * NN;  // A row per lane
    const float* eb   = emb_in + (size_t)b * NN * EP;

    v8f acc = {0.f, 0.f, 0.f, 0.f, 0.f, 0.f, 0.f, 0.f};

    const int kbeg = wave * (NN / 8);              // split-K across the 8 waves
    for (int k = kbeg; k < kbeg + NN / 8; k += 4) {
        const int k2 = k + khalf * 2;
        // A fragment: 16x4 f32 layout — lanes 0-15 hold K={k,k+1}, lanes 16-31 K={k+2,k+3}
        v2f a;
        a.x = arow[k2];
        a.y = arow[k2 + 1];
        // B fragment: 4x16 f32 — row k striped across 16 lanes per VGPR
        v2f bf;
        bf.x = eb[(size_t)k2 * EP + col];
        bf.y = eb[(size_t)(k2 + 1) * EP + col];
        acc = __builtin_amdgcn_wmma_f32_16x16x4_f32(
                  false, a, false, bf, (short)0, acc, false, false);
    }

    // C/D layout: VGPR r -> row (khalf*8 + r), col = lane&15
    const int rb = khalf * 8;
    #pragma unroll
    for (int r = 0; r < 8; ++r)
        sP[wave * 256 + (rb + r) * 16 + col] = acc[r];
    __syncthreads();

    // Reduce the 8 partial tiles: thread t owns element (t>>4, t&15).
    float s = 0.f;
    #pragma unroll
    for (int w = 0; w < 8; ++w) s += sP[w * 256 + threadIdx.x];
    __syncthreads();
    sP[threadIdx.x] = s;                           // reduced tile P[16][16]
    __syncthreads();

    // Epilogue: out[row,o] = relu(base + nbp_b + sum_e P[row,e]*nbp_w[o,e])
    const int pr = threadIdx.x >> 4;               // tile row
    const int pc = threadIdx.x & 15;               // output column o
    const int gRow = i0 + pr;
    const size_t o0 = ((size_t)b * NN + gRow) * EP;

    float v = 0.f;
    if (pc < EE) {
        v = base[o0 + pc] + nbp_b[pc];
        #pragma unroll
        for (int e = 0; e < EE; ++e) v = fmaf(sP[pr * 16 + e], nbp_w[pc * EE + e], v);
        v = (v > 0.f) ? v : 0.f;
    }
    emb_out[o0 + pc] = v;                          // keeps padding at 0
}

// ---------------------------------------------------------------------------
// Kernel 3: readout.  q[b,n] = c_b + dot(emb[b,n,:], r)  where
//   r[e]  = sum_o qred_w[E+o]*qact_w[o,e]
//   c_b   = qred_b + sum_o qred_w[o]*sumembed[b,o] + sum_o qred_w[E+o]*qact_b[o]
// Also emits the unpadded emb output.
// ---------------------------------------------------------------------------
__global__ void k_readout(const float* __restrict__ emb,     // padded [B][N][EP]
                          const float* __restrict__ qred_w, const float* __restrict__ qred_b,
                          const float* __restrict__ qall_w, const float* __restrict__ qall_b,
                          const float* __restrict__ qact_w, const float* __restrict__ qact_b,
                          float* __restrict__ q_out, float* __restrict__ emb_out)
{
    __shared__ float sred[256];
    __shared__ float scol[EE];
    __shared__ float sc[1 + EE];                   // [0]=c_b, [1..]=r[e]

    const int b = blockIdx.x;
    const int t = threadIdx.x;
    const float* eb = emb + (size_t)b * NN * EP;

    float part[EE];
    #pragma unroll
    for (int e = 0; e < EE; ++e) part[e] = 0.f;
    for (int n = t; n < NN; n += 256) {
        #pragma unroll
        for (int e = 0; e < EE; ++e) part[e] += eb[(size_t)n * EP + e];
    }
    for (int e = 0; e < EE; ++e) {
        sred[t] = part[e];
        __syncthreads();
        for (int s = 128; s > 0; s >>= 1) {
            if (t < s) sred[t] += sred[t + s];
            __syncthreads();
        }
        if (t == 0) scol[e] = sred[0];
        __syncthreads();
    }

    if (t == 0) {
        float cb = qred_b[0];
        for (int o = 0; o < EE; ++o) {
            float se = qall_b[o];
            for (int e = 0; e < EE; ++e) se = fmaf(scol[e], qall_w[o * EE + e], se);
            cb = fmaf(qred_w[o], se, cb);
            cb = fmaf(qred_w[EE + o], qact_b[o], cb);
        }
        sc[0] = cb;
        for (int e = 0; e < EE; ++e) {
            float r = 0.f;
            for (int o = 0; o < EE; ++o) r = fmaf(qred_w[EE + o], qact_w[o * EE + e], r);
            sc[1 + e] = r;
        }
    }
    __syncthreads();

    const float cb = sc[0];
    for (int n = t; n < NN; n += 256) {
        float q = cb;
        #pragma unroll
        for (int e = 0; e < EE; ++e) {
            const float ev = eb[(size_t)n * EP + e];
            q = fmaf(ev, sc[1 + e], q);
            emb_out[((size_t)b * NN + n) * EE + e] = ev;
        }
        q_out[b * NN + n] = q;
    }
}

// ---------------------------------------------------------------------------
extern "C" void kernel_launch(void* const* d_in, const int* in_sizes, int n_in,
                              void* d_out, int out_size, void* d_ws, size_t ws_size,
                              hipStream_t stream)
{
    const float* features  = (const float*)d_in[0];
    const float* weights   = (const float*)d_in[1];
    const float* adjacency = (const float*)d_in[2];
    const float* sel_w   = (const float*)d_in[3];
    const float* sel_b   = (const float*)d_in[4];
    const float* nbp_w   = (const float*)d_in[5];
    const float* nbp_b   = (const float*)d_in[6];
    const float* nbw_w   = (const float*)d_in[7];
    const float* nbw_b   = (const float*)d_in[8];
    const float* nbwew_w = (const float*)d_in[9];
    const float* nbwew_b = (const float*)d_in[10];
    const float* qred_w  = (const float*)d_in[11];
    const float* qred_b  = (const float*)d_in[12];
    const float* qall_w  = (const float*)d_in[13];
    const float* qall_b  = (const float*)d_in[14];
    const float* qact_w  = (const float*)d_in[15];
    const float* qact_b  = (const float*)d_in[16];

    float* base = (float*)d_ws;                       // [B][N][EP]
    float* embA = base + (size_t)BB * NN * EP;        // [B][N][EP]
    float* embB = embA + (size_t)BB * NN * EP;        // [B][N][EP]

    // Stage 1: edge-weight reduction + base + iteration-1 emb.  256 blocks.
    k_edge_base<<<dim3(NN / 32, BB), 256, 0, stream>>>(
        features, weights, sel_w, sel_b, nbw_w, nbw_b, nbwew_w, nbwew_b, nbp_b,
        base, embA);

    // Stage 2: iterations 2 and 3 (WMMA adjacency matmul, split-K=8 per
    // block), ping-pong buffers.  512 blocks each.
    k_propagate<<<dim3(NN / 16, BB), 256, 0, stream>>>(
        adjacency, base, nbp_w, nbp_b, embA, embB);
    k_propagate<<<dim3(NN / 16, BB), 256, 0, stream>>>(
        adjacency, base, nbp_w, nbp_b, embB, embA);

    // Stage 3: readout.  d_out = q_vals [B*N] ++ emb [B*N*E]
    float* q_out   = (float*)d_out;
    float* emb_out = q_out + (size_t)BB * NN;
    k_readout<<<BB, 256, 0, stream>>>(
        embA, qred_w, qred_b, qall_w, qall_b, qact_w, qact_b, q_out, emb_out);
}